// OhemCrossEntropyLoss2d_6614249636064
// MI455X (gfx1250) — compile-verified
//
#include <hip/hip_runtime.h>
#include <stdint.h>

// OHEM cross-entropy: fused CE + histogram-based radix selection instead of a
// full 4.19M-element sort. All order statistics computed exactly on float bit
// patterns (losses are non-negative => uint bit order == value order).

#define P_TOTAL   4194304u            // N*H*W = 8*512*1024
#define HW        524288u             // 512*1024 (power of two -> shifts)
#define C_DIM     19
#define CHW       ((size_t)C_DIM * HW)
#define THRESH_F  0.35667494393873245f  // -log(0.7)
#define MIN_KEPT  100000u
#define K_RANK    100001u             // 1-indexed rank of sorted_desc[100000]
#define IGNORE_IDX 255
#define NBLK      1024
#define NTHR      256                 // 8 wave32 waves per block
#define CHUNK     (NTHR * 4)          // 1024 floats per async-staged chunk
#define NCHUNK    (P_TOTAL / CHUNK)   // 4096 chunks (exact)

typedef unsigned long long ull;

__device__ __forceinline__ void blockReduce(double &s, ull &c,
                                            double* shS, ull* shC) {
  const int tid = threadIdx.x;
  shS[tid] = s; shC[tid] = c;
  __syncthreads();
  for (int off = NTHR / 2; off > 0; off >>= 1) {
    if (tid < off) { shS[tid] += shS[tid + off]; shC[tid] += shC[tid + off]; }
    __syncthreads();
  }
  s = shS[0]; c = shC[0];
}

// Low 32 bits of a flat shared-memory address == LDS byte offset
// (ISA: LDS aperture truncates ADDR[31:0]).
__device__ __forceinline__ unsigned lds_off_of(const void* p) {
  return (unsigned)(uintptr_t)p;
}

// CDNA5 async copy engine: global -> LDS, 16 bytes per lane, ASYNCcnt-tracked.
__device__ __forceinline__ void async_copy_b128(unsigned lds_off,
                                                const float* gaddr) {
  asm volatile("global_load_async_to_lds_b128 %0, %1, off"
               :: "v"(lds_off), "v"(gaddr)
               : "memory");
}

// Pass 1: fused log-softmax NLL, store per-pixel loss, 2048-bin LDS histogram
// of the top 11 float bits, and partial sum/count of losses > THRESH.
__global__ void loss_hist_kernel(const float* __restrict__ logits,
                                 const int* __restrict__ labels,
                                 float* __restrict__ loss,
                                 unsigned* __restrict__ hist,
                                 double* __restrict__ pSumT,
                                 ull* __restrict__ pCntT) {
  __shared__ unsigned shH[2048];
  __shared__ double shS[NTHR];
  __shared__ ull shC[NTHR];
  const int tid = threadIdx.x;
  for (int i = tid; i < 2048; i += NTHR) shH[i] = 0;
  __syncthreads();

  double s = 0.0; ull c = 0;
  for (unsigned p = blockIdx.x * NTHR + tid; p < P_TOTAL; p += NBLK * NTHR) {
    const unsigned n  = p >> 19;          // p / HW
    const unsigned hw = p & (HW - 1u);    // p % HW
    const float* px = logits + (size_t)n * CHW + hw;
    // gfx1250 speculative prefetch for the next grid-stride iteration
    __builtin_prefetch(px + (size_t)(NBLK * NTHR), 0, 0);

    float v[C_DIM];
#pragma unroll
    for (int ch = 0; ch < C_DIM; ++ch) v[ch] = px[(size_t)ch * HW];

    float mx = v[0];
#pragma unroll
    for (int ch = 1; ch < C_DIM; ++ch) mx = fmaxf(mx, v[ch]);
    float se = 0.0f;
#pragma unroll
    for (int ch = 0; ch < C_DIM; ++ch) se += __expf(v[ch] - mx);
    const float lse = __logf(se) + mx;

    const int lbl = labels[p];
    const int cl  = min(max(lbl, 0), C_DIM - 1);
    const float l = (lbl != IGNORE_IDX) ? (lse - v[cl]) : 0.0f;

    loss[p] = l;
    atomicAdd(&shH[__float_as_uint(l) >> 21], 1u);
    if (l > THRESH_F) { s += (double)l; ++c; }
  }
  __syncthreads();
  for (int i = tid; i < 2048; i += NTHR) {
    const unsigned h = shH[i];
    if (h) atomicAdd(&hist[i], h);
  }
  blockReduce(s, c, shS, shC);
  if (tid == 0) { pSumT[blockIdx.x] = s; pCntT[blockIdx.x] = c; }
}

// Descending scan of the histogram to locate the bin containing rank remK.
// state[0]=prefix bits found so far, state[1]=remaining rank, state[2]=kth bits.
template <int LEVEL>
__global__ void scan_kernel(unsigned* __restrict__ hist,
                            unsigned* __restrict__ state) {
  if (threadIdx.x == 0) {
    unsigned remK = (LEVEL == 0) ? K_RANK : state[1];
    const int nb = (LEVEL == 2) ? 1024 : 2048;
    ull cum = 0;
    int found = 0;
    for (int b = nb - 1; b >= 0; --b) {
      const unsigned h = hist[b];
      if (cum + h >= remK) { found = b; state[1] = remK - (unsigned)cum; break; }
      cum += h;
    }
    if (LEVEL == 0)      state[0] = (unsigned)found << 21;
    else if (LEVEL == 1) state[0] |= (unsigned)found << 10;
    else                 state[2] = state[0] | (unsigned)found;
  }
  __syncthreads();
  for (int i = threadIdx.x; i < 2048; i += blockDim.x) hist[i] = 0;
}

// Refinement passes: histogram the next digit of elements matching the prefix.
template <int LEVEL>
__global__ void pass_kernel(const float* __restrict__ loss,
                            unsigned* __restrict__ hist,
                            const unsigned* __restrict__ state) {
  const unsigned prefix = state[0];
  for (unsigned p = blockIdx.x * NTHR + threadIdx.x; p < P_TOTAL;
       p += NBLK * NTHR) {
    const unsigned u = __float_as_uint(loss[p]);
    if (LEVEL == 1) {
      if ((u >> 21) == (prefix >> 21)) atomicAdd(&hist[(u >> 10) & 2047u], 1u);
    } else {
      if ((u >> 10) == (prefix >> 10)) atomicAdd(&hist[u & 1023u], 1u);
    }
  }
}

// Exact sum/count of elements strictly greater than the k-th value.
// Streams the loss array through LDS with double-buffered async copies:
// each thread owns a private 16B slot per stage, so only ASYNCcnt waits are
// needed (async loads complete in order within a wave) -- no barriers.
__global__ void count_kernel(const float* __restrict__ loss,
                             const unsigned* __restrict__ state,
                             double* __restrict__ pSumG,
                             ull* __restrict__ pCntG) {
  __shared__ __align__(16) float buf[2][CHUNK];
  __shared__ double shS[NTHR];
  __shared__ ull shC[NTHR];
  const int tid = threadIdx.x;
  const unsigned kb = state[2];
  const unsigned slot = (unsigned)tid * 4u;
  const unsigned ldsA = lds_off_of(&buf[0][slot]);
  const unsigned ldsB = lds_off_of(&buf[1][slot]);

  double s = 0.0; ull c = 0;
  const unsigned nloc = NCHUNK / NBLK;          // 4 chunks per block (exact)

  // prologue: chunk 0 -> stage 0
  async_copy_b128(ldsA, loss + (size_t)blockIdx.x * CHUNK + slot);
  for (unsigned i = 0; i < nloc; ++i) {
    if (i + 1 < nloc) {
      const unsigned nxt = blockIdx.x + (i + 1) * NBLK;
      async_copy_b128((i & 1) ? ldsA : ldsB,      // stage (i+1)&1
                      loss + (size_t)nxt * CHUNK + slot);
      asm volatile("s_wait_asynccnt 0x1" ::: "memory");  // oldest copy done
    } else {
      asm volatile("s_wait_asynccnt 0x0" ::: "memory");  // drain
    }
    const float4 q = *(const float4*)&buf[i & 1][slot];  // ds_load_b128
    if (__float_as_uint(q.x) > kb) { s += (double)q.x; ++c; }
    if (__float_as_uint(q.y) > kb) { s += (double)q.y; ++c; }
    if (__float_as_uint(q.z) > kb) { s += (double)q.z; ++c; }
    if (__float_as_uint(q.w) > kb) { s += (double)q.w; ++c; }
  }
  blockReduce(s, c, shS, shC);
  if (tid == 0) { pSumG[blockIdx.x] = s; pCntG[blockIdx.x] = c; }
}

__global__ void final_kernel(const unsigned* __restrict__ state,
                             const double* __restrict__ pSumT,
                             const ull* __restrict__ pCntT,
                             const double* __restrict__ pSumG,
                             const ull* __restrict__ pCntG,
                             float* __restrict__ out) {
  __shared__ double shA[NTHR], shB[NTHR];
  __shared__ ull shC[NTHR], shD[NTHR];
  const int tid = threadIdx.x;
  double sT = 0.0, sG = 0.0; ull cT = 0, cG = 0;
  for (int i = tid; i < NBLK; i += NTHR) {
    sT += pSumT[i]; cT += pCntT[i]; sG += pSumG[i]; cG += pCntG[i];
  }
  shA[tid] = sT; shB[tid] = sG; shC[tid] = cT; shD[tid] = cG;
  __syncthreads();
  for (int off = NTHR / 2; off > 0; off >>= 1) {
    if (tid < off) {
      shA[tid] += shA[tid + off]; shB[tid] += shB[tid + off];
      shC[tid] += shC[tid + off]; shD[tid] += shD[tid + off];
    }
    __syncthreads();
  }
  if (tid == 0) {
    const float v = __uint_as_float(state[2]);      // sorted_desc[100000]
    const double meanT = shA[0] / (double)shC[0];
    // top-100k = all elements > v plus (100000 - count_greater) copies of v
    const double topk =
        (shB[0] + (double)(MIN_KEPT - shD[0]) * (double)v) / (double)MIN_KEPT;
    out[0] = (v > THRESH_F) ? (float)meanT : (float)topk;
  }
}

extern "C" void kernel_launch(void* const* d_in, const int* in_sizes, int n_in,
                              void* d_out, int out_size, void* d_ws,
                              size_t ws_size, hipStream_t stream) {
  const float* logits = (const float*)d_in[0];
  const int*   labels = (const int*)d_in[1];
  float* out = (float*)d_out;

  uint8_t* ws = (uint8_t*)d_ws;
  float* loss = (float*)ws;
  size_t off = (size_t)P_TOTAL * sizeof(float);        // 16 MB loss buffer
  unsigned* hist  = (unsigned*)(ws + off); off += 2048 * sizeof(unsigned);
  unsigned* state = (unsigned*)(ws + off); off += 64;  // prefix/remK/kthBits
  double* pSumT = (double*)(ws + off);     off += NBLK * sizeof(double);
  ull*    pCntT = (ull*)(ws + off);        off += NBLK * sizeof(ull);
  double* pSumG = (double*)(ws + off);     off += NBLK * sizeof(double);
  ull*    pCntG = (ull*)(ws + off);        off += NBLK * sizeof(ull);

  // hist + state are contiguous; must be zeroed every call (capture-safe).
  hipMemsetAsync(hist, 0, 2048 * sizeof(unsigned) + 64, stream);

  loss_hist_kernel<<<NBLK, NTHR, 0, stream>>>(logits, labels, loss, hist,
                                              pSumT, pCntT);
  scan_kernel<0><<<1, NTHR, 0, stream>>>(hist, state);
  pass_kernel<1><<<NBLK, NTHR, 0, stream>>>(loss, hist, state);
  scan_kernel<1><<<1, NTHR, 0, stream>>>(hist, state);
  pass_kernel<2><<<NBLK, NTHR, 0, stream>>>(loss, hist, state);
  scan_kernel<2><<<1, NTHR, 0, stream>>>(hist, state);
  count_kernel<<<NBLK, NTHR, 0, stream>>>(loss, state, pSumG, pCntG);
  final_kernel<<<1, NTHR, 0, stream>>>(state, pSumT, pCntT, pSumG, pCntG, out);
}